// AbsoluteDecoderBlock_28166395528092
// MI455X (gfx1250) — compile-verified
//
#include <hip/hip_runtime.h>
#include <hip/hip_bf16.h>
#include <math.h>

typedef float v2f __attribute__((ext_vector_type(2)));
typedef float v4f __attribute__((ext_vector_type(4)));
typedef float v8f __attribute__((ext_vector_type(8)));

#define NBATCH 16
#define CCH    256
#define HDIM   64
#define WDIM   64
#define DDIM   64
#define KCB    512
#define NPIX   (NBATCH*HDIM*WDIM)   /* 65536 */
#define C3     (3*CCH)              /* 768 */
#define EPSBN  1e-5f

// ---- masked-conv tap tables ----
// mask A (3x3): keeps (ky,kx) = (0,0)(0,1)(0,2)(1,0); offsets oy=ky-1, ox=kx-1
__constant__ int c_toyA[4] = {-1,-1,-1, 0};
__constant__ int c_toxA[4] = {-1, 0, 1,-1};
__constant__ int c_kyA[4]  = { 0, 0, 0, 1};
__constant__ int c_kxA[4]  = { 0, 1, 2, 0};
// mask B (5x5): rows 0,1 full; row 2 cols 0..2; offsets oy=ky-2, ox=kx-2
__constant__ int c_toyB[13] = {-2,-2,-2,-2,-2,-1,-1,-1,-1,-1, 0, 0, 0};
__constant__ int c_toxB[13] = {-2,-1, 0, 1, 2,-2,-1, 0, 1, 2,-2,-1, 0};
__constant__ int c_kyB[13]  = { 0, 0, 0, 0, 0, 1, 1, 1, 1, 1, 2, 2, 2};
__constant__ int c_kxB[13]  = { 0, 1, 2, 3, 4, 0, 1, 2, 3, 4, 0, 1, 2};

// =====================================================================
// Layout transforms
// =====================================================================
__global__ void k_nchw2nhwc(const float* __restrict__ src, float* __restrict__ dst) {
  size_t i = (size_t)blockIdx.x * blockDim.x + threadIdx.x;   // over N*C (NCHW flat)
  int b = (int)(i >> 20);            // C*H*W = 2^20
  int c = (int)((i >> 12) & 255);    // H*W   = 2^12
  int p = (int)(i & 4095);
  dst[((size_t)b * 4096 + p) * CCH + c] = src[i];
}

// pack masked weights tap-major: wp[t][oc][ic]
__global__ void k_pack_wm1(const float* __restrict__ w, float* __restrict__ wp) {
  int i = blockIdx.x * blockDim.x + threadIdx.x;      // 4*64*64
  int t = i / (DDIM * DDIM);
  int rem = i - t * DDIM * DDIM;
  int oc = rem / DDIM, ic = rem - oc * DDIM;
  wp[i] = w[((size_t)oc * DDIM + ic) * 9 + c_kyA[t] * 3 + c_kxA[t]];
}
__global__ void k_pack_wm2(const float* __restrict__ w, float* __restrict__ wp) {
  int i = blockIdx.x * blockDim.x + threadIdx.x;      // 13*512*64
  int t = i / (KCB * DDIM);
  int rem = i - t * KCB * DDIM;
  int oc = rem / DDIM, ic = rem - oc * DDIM;
  wp[i] = w[((size_t)oc * DDIM + ic) * 25 + c_kyB[t] * 5 + c_kxB[t]];
}
__global__ void k_ek2(const float* __restrict__ emb, float* __restrict__ ek2) {
  int k = blockIdx.x * blockDim.x + threadIdx.x;      // 512
  float s = 0.f;
  for (int d = 0; d < DDIM; ++d) { float v = emb[(size_t)k * DDIM + d]; s += v * v; }
  ek2[k] = s;
}

// =====================================================================
// Generic WMMA f32 GEMM:  Y[N, OCH] = A[N, KD] * Bw[OCH, KD]^T + bias
// Optional BN on A operand (per-k mean / rsigma).
// One wave -> 16 rows x 64 cols (4 accumulator tiles): the per-lane A
// float2 is loaded once per k-step and reused by 4 WMMAs (operand
// traffic 1.25 loads/WMMA instead of 2).
// =====================================================================
template<int KD, int OCH, bool BN>
__global__ void k_gemm(const float* __restrict__ A, const float* __restrict__ Bw,
                       const float* __restrict__ bias, float* __restrict__ Y,
                       const float* __restrict__ mean, const float* __restrict__ rs) {
  const int colGroups = OCH / 64;
  int wave = blockIdx.x * (blockDim.x >> 5) + (threadIdx.x >> 5);
  if (wave >= (NPIX / 16) * colGroups) return;          // wave-uniform
  int tn = wave / colGroups, tg = wave - tn * colGroups;
  int row0 = tn * 16, col0 = tg * 64;
  int lane = threadIdx.x & 31, half = lane >> 4, m = lane & 15;
  const float* ar = A + (size_t)(row0 + m) * KD + 2 * half;
  const float* br[4];
  for (int j = 0; j < 4; ++j) {
    br[j] = Bw + (size_t)(col0 + 16 * j + m) * KD + 2 * half;
    __builtin_prefetch(br[j], 0, 1);                    // global_prefetch_b8
  }
  v8f acc[4] = {{}, {}, {}, {}};
  for (int k = 0; k < KD; k += 4) {
    v2f a = *(const v2f*)(ar + k);
    if constexpr (BN) {
      v2f mm = *(const v2f*)(mean + k + 2 * half);
      v2f rr = *(const v2f*)(rs   + k + 2 * half);
      a = (a - mm) * rr;
    }
    for (int j = 0; j < 4; ++j) {
      v2f b = *(const v2f*)(br[j] + k);
      acc[j] = __builtin_amdgcn_wmma_f32_16x16x4_f32(false, a, false, b, (short)0,
                                                     acc[j], false, false);
    }
  }
  for (int j = 0; j < 4; ++j) {
    float bb = bias[col0 + 16 * j + m];
    for (int r = 0; r < 8; ++r)
      Y[(size_t)(row0 + r + 8 * half) * OCH + col0 + 16 * j + m] = acc[j][r] + bb;
  }
}

// =====================================================================
// Masked conv as tap-wise implicit WMMA GEMM. X: [N,64] NHWC, Wp: [T,OCH,64]
// 16-pixel tiles never cross an image row (W=64, tiles of 16).
// Same 4-way column blocking: per-tap A loads reused by 4 WMMAs.
// =====================================================================
template<int NTAPS, int OCH>
__global__ void k_mconv(const float* __restrict__ X, const float* __restrict__ Wp,
                        const float* __restrict__ bias, float* __restrict__ Y,
                        const float* __restrict__ zrow) {
  const int colGroups = OCH / 64;
  int wave = blockIdx.x * (blockDim.x >> 5) + (threadIdx.x >> 5);
  if (wave >= (NPIX / 16) * colGroups) return;
  int tn = wave / colGroups, tg = wave - tn * colGroups;
  int row0 = tn * 16, col0 = tg * 64;
  int lane = threadIdx.x & 31, half = lane >> 4, m = lane & 15;
  int b = row0 >> 12, h = (row0 >> 6) & 63, wl = (row0 & 63) + m;
  const int* toy = (NTAPS == 4) ? c_toyA : c_toyB;
  const int* tox = (NTAPS == 4) ? c_toxA : c_toxB;
  v8f acc[4] = {{}, {}, {}, {}};
  for (int t = 0; t < NTAPS; ++t) {
    int hh = h + toy[t];
    int ww = wl + tox[t];
    bool valid = ((unsigned)hh < 64u) && ((unsigned)ww < 64u);
    const float* ar = valid ? (X + ((size_t)(((b << 6) + hh) << 6) + ww) * DDIM + 2 * half)
                            : zrow;
    const float* br[4];
    for (int j = 0; j < 4; ++j) {
      br[j] = Wp + ((size_t)t * OCH + col0 + 16 * j + m) * DDIM + 2 * half;
      __builtin_prefetch(br[j], 0, 1);
    }
    for (int k = 0; k < DDIM; k += 4) {
      v2f a = *(const v2f*)(ar + k);
      for (int j = 0; j < 4; ++j) {
        v2f bv = *(const v2f*)(br[j] + k);
        acc[j] = __builtin_amdgcn_wmma_f32_16x16x4_f32(false, a, false, bv, (short)0,
                                                       acc[j], false, false);
      }
    }
  }
  for (int j = 0; j < 4; ++j) {
    float bb = bias[col0 + 16 * j + m];
    for (int r = 0; r < 8; ++r)
      Y[(size_t)(row0 + r + 8 * half) * OCH + col0 + 16 * j + m] = acc[j][r] + bb;
  }
}

// =====================================================================
// VQ argmin: score(n,k) = ||e_k||^2 - 2 z_n . e_k  via WMMA tiles over K
// =====================================================================
__global__ void k_vq(const float* __restrict__ zf, const float* __restrict__ emb,
                     const float* __restrict__ ek2, int* __restrict__ idx) {
  int wave = blockIdx.x * (blockDim.x >> 5) + (threadIdx.x >> 5);
  if (wave >= NPIX / 16) return;
  int row0 = wave * 16;
  int lane = threadIdx.x & 31, half = lane >> 4, m = lane & 15;
  const float* ar = zf + (size_t)(row0 + m) * DDIM + 2 * half;
  float minv[8]; int mini[8];
  for (int r = 0; r < 8; ++r) { minv[r] = 3.4e38f; mini[r] = 0; }
  for (int kt = 0; kt < KCB / 16; ++kt) {
    int col = kt * 16 + m;
    const float* br = emb + (size_t)col * DDIM + 2 * half;
    v8f acc = {};
    for (int k = 0; k < DDIM; k += 4)
      acc = __builtin_amdgcn_wmma_f32_16x16x4_f32(false, *(const v2f*)(ar + k),
                                                  false, *(const v2f*)(br + k),
                                                  (short)0, acc, false, false);
    float e2 = ek2[col];
    for (int r = 0; r < 8; ++r) {
      float s = e2 - 2.0f * acc[r];
      if (s < minv[r] || (s == minv[r] && col < mini[r])) { minv[r] = s; mini[r] = col; }
    }
  }
  for (int off = 1; off < 16; off <<= 1) {
    for (int r = 0; r < 8; ++r) {
      float ov = __shfl_xor(minv[r], off, 32);
      int   oi = __shfl_xor(mini[r], off, 32);
      if (ov < minv[r] || (ov == minv[r] && oi < mini[r])) { minv[r] = ov; mini[r] = oi; }
    }
  }
  if (m == 0)
    for (int r = 0; r < 8; ++r) idx[row0 + r + 8 * half] = mini[r];
}

__global__ void k_hist(const int* __restrict__ idx, int* __restrict__ counts) {
  int n = blockIdx.x * blockDim.x + threadIdx.x;
  atomicAdd(&counts[idx[n]], 1);
}

// gather codes (straight-through forward value) + commitment partial sums
__global__ void k_gather_commit(const float* __restrict__ zf, const float* __restrict__ emb,
                                const int* __restrict__ idx, float* __restrict__ qN,
                                float* __restrict__ partials) {
  __shared__ float sh[256];
  size_t i = (size_t)blockIdx.x * 256 + threadIdx.x;   // over N*D
  int n = (int)(i >> 6), d = (int)(i & 63);
  float q = emb[(size_t)idx[n] * DDIM + d];
  qN[i] = q;
  float dd = q - zf[i];
  sh[threadIdx.x] = dd * dd;
  __syncthreads();
  for (int st = 128; st; st >>= 1) {
    if (threadIdx.x < st) sh[threadIdx.x] += sh[threadIdx.x + st];
    __syncthreads();
  }
  if (threadIdx.x == 0) partials[blockIdx.x] = sh[0];
}

__global__ void k_combine(const float* __restrict__ partials, int n, float scale,
                          float* __restrict__ out) {
  __shared__ float sh[256];
  float s = 0.f;
  for (int i = threadIdx.x; i < n; i += 256) s += partials[i];
  sh[threadIdx.x] = s;
  __syncthreads();
  for (int st = 128; st; st >>= 1) {
    if (threadIdx.x < st) sh[threadIdx.x] += sh[threadIdx.x + st];
    __syncthreads();
  }
  if (threadIdx.x == 0) *out = sh[0] * scale;
}

// =====================================================================
// BatchNorm (train mode): deterministic per-channel stats, then apply
// =====================================================================
__global__ void k_bnstats(const float* __restrict__ X, int Cdim,
                          float* __restrict__ mean, float* __restrict__ rs) {
  int c = blockIdx.x;
  double s = 0.0, s2 = 0.0;
  for (int n = threadIdx.x; n < NPIX; n += 256) {
    float v = X[(size_t)n * Cdim + c];
    s += v; s2 += (double)v * v;
  }
  __shared__ double shs[256], shq[256];
  shs[threadIdx.x] = s; shq[threadIdx.x] = s2;
  __syncthreads();
  for (int st = 128; st; st >>= 1) {
    if (threadIdx.x < st) { shs[threadIdx.x] += shs[threadIdx.x + st];
                            shq[threadIdx.x] += shq[threadIdx.x + st]; }
    __syncthreads();
  }
  if (threadIdx.x == 0) {
    double mu = shs[0] / NPIX;
    double var = shq[0] / NPIX - mu * mu;
    mean[c] = (float)mu;
    rs[c] = (float)(1.0 / sqrt(var + (double)EPSBN));
  }
}

__global__ void k_bnapply(float* __restrict__ X, int Cdim, const float* __restrict__ mean,
                          const float* __restrict__ rs, int doSwish) {
  size_t i = (size_t)blockIdx.x * blockDim.x + threadIdx.x;
  int c = (int)(i % Cdim);
  float v = (X[i] - mean[c]) * rs[c];
  if (doSwish) v = v / (1.f + expf(-v));
  X[i] = v;
}

// =====================================================================
// Depthwise 5x5, pad 2, 768 channels; 4 channels per thread (float4)
// =====================================================================
__global__ void k_dw(const float* __restrict__ hA, const float* __restrict__ wdw,
                     const float* __restrict__ bdw, float* __restrict__ hB) {
  int t = blockIdx.x * blockDim.x + threadIdx.x;    // NPIX * 192
  int n = t / 192, cg = t - n * 192, c0 = cg * 4;
  int b = n >> 12, h = (n >> 6) & 63, w = n & 63;
  v4f acc = { bdw[c0], bdw[c0 + 1], bdw[c0 + 2], bdw[c0 + 3] };
  for (int ky = 0; ky < 5; ++ky) {
    int hh = h + ky - 2;
    if ((unsigned)hh >= 64u) continue;
    for (int kx = 0; kx < 5; ++kx) {
      int ww = w + kx - 2;
      if ((unsigned)ww >= 64u) continue;
      const float* p = hA + ((size_t)(((b << 6) + hh) << 6) + ww) * C3 + c0;
      v4f a = *(const v4f*)p;
      int tap = ky * 5 + kx;
      v4f wv = { wdw[(c0    ) * 25 + tap], wdw[(c0 + 1) * 25 + tap],
                 wdw[(c0 + 2) * 25 + tap], wdw[(c0 + 3) * 25 + tap] };
      acc += a * wv;
    }
  }
  *(v4f*)(hB + (size_t)n * C3 + c0) = acc;
}

// =====================================================================
// Prior NLL: per-row logsumexp minus target logit (one wave per row)
// =====================================================================
__global__ void k_nll(const float* __restrict__ logits, const int* __restrict__ idx,
                      float* __restrict__ partials) {
  __shared__ float ssum[8];
  int waveId = threadIdx.x >> 5, lane = threadIdx.x & 31;
  int row = blockIdx.x * 8 + waveId;
  const float* lr = logits + (size_t)row * KCB;
  float v[16], mx = -3.4e38f;
  for (int j = 0; j < 16; ++j) { v[j] = lr[lane + 32 * j]; mx = fmaxf(mx, v[j]); }
  for (int off = 16; off; off >>= 1) mx = fmaxf(mx, __shfl_xor(mx, off, 32));
  float se = 0.f;
  for (int j = 0; j < 16; ++j) se += expf(v[j] - mx);
  for (int off = 16; off; off >>= 1) se += __shfl_xor(se, off, 32);
  float lse = mx + logf(se);
  if (lane == 0) ssum[waveId] = lse - lr[idx[row]];
  __syncthreads();
  if (threadIdx.x == 0) {
    float s = 0.f;
    for (int i = 0; i < 8; ++i) s += ssum[i];
    partials[blockIdx.x] = s;
  }
}

// =====================================================================
// Squeeze-excite
// =====================================================================
__global__ void k_squeeze(const float* __restrict__ hOut, float* __restrict__ sbuf) {
  int bc = blockIdx.x;                    // 16*256
  int b = bc >> 8, c = bc & 255;
  float s = 0.f;
  for (int p = threadIdx.x; p < 4096; p += 128)
    s += hOut[((size_t)b * 4096 + p) * CCH + c];
  __shared__ float sh[128];
  sh[threadIdx.x] = s;
  __syncthreads();
  for (int st = 64; st; st >>= 1) {
    if (threadIdx.x < st) sh[threadIdx.x] += sh[threadIdx.x + st];
    __syncthreads();
  }
  if (threadIdx.x == 0) sbuf[bc] = sh[0] * (1.f / 4096.f);
}

__global__ void k_se(const float* __restrict__ sbuf,
                     const float* __restrict__ w1, const float* __restrict__ b1,
                     const float* __restrict__ w2, const float* __restrict__ b2,
                     float* __restrict__ s2buf) {
  __shared__ float s1s[256];              // [16 batches][16 hidden]
  int t = threadIdx.x;
  {
    int bi = t >> 4, j = t & 15;
    float s = b1[j];
    for (int c = 0; c < CCH; ++c) s += sbuf[bi * CCH + c] * w1[j * CCH + c];
    s1s[t] = fmaxf(s, 0.f);
  }
  __syncthreads();
  for (int e = t; e < NBATCH * CCH; e += 256) {
    int bi = e >> 8, c = e & 255;
    float s = b2[c];
    for (int j = 0; j < 16; ++j) s += s1s[bi * 16 + j] * w2[c * 16 + j];
    s2buf[e] = 1.f / (1.f + expf(-s));
  }
}

__global__ void k_final(const float* __restrict__ xN, const float* __restrict__ hOut,
                        const float* __restrict__ s2, float* __restrict__ out) {
  size_t o = (size_t)blockIdx.x * blockDim.x + threadIdx.x;   // NCHW flat
  int b = (int)(o >> 20), c = (int)((o >> 12) & 255), p = (int)(o & 4095);
  size_t n = (size_t)b * 4096 + p;
  out[o] = xN[n * CCH + c] + hOut[n * CCH + c] * s2[b * CCH + c];
}

__global__ void k_scalars(const float* __restrict__ accum, const int* __restrict__ counts,
                          float* __restrict__ out) {
  if (threadIdx.x != 0 || blockIdx.x != 0) return;
  float ent = 0.f; int used = 0;
  for (int k = 0; k < KCB; ++k) {
    float p = (float)counts[k] / (float)NPIX;
    ent += p * logf(p + 1e-10f);
    if (counts[k] > 0) ++used;
  }
  out[0] = accum[0];                 // commitment loss (already / (N*D))
  out[1] = accum[1];                 // sample loss    (already / N)
  out[2] = expf(-ent);               // perplexity
  out[3] = (float)used / (float)KCB; // usage
}

// =====================================================================
extern "C" void kernel_launch(void* const* d_in, const int* in_sizes, int n_in,
                              void* d_out, int out_size, void* d_ws, size_t ws_size,
                              hipStream_t stream) {
  const float* feature = (const float*)d_in[0];
  const float* emb     = (const float*)d_in[1];
  const float* w_q  = (const float*)d_in[2];   const float* b_q  = (const float*)d_in[3];
  const float* w_c  = (const float*)d_in[4];   const float* b_c  = (const float*)d_in[5];
  const float* w_d1 = (const float*)d_in[6];   const float* b_d1 = (const float*)d_in[7];
  const float* w_dw = (const float*)d_in[8];   const float* b_dw = (const float*)d_in[9];
  const float* w_d2 = (const float*)d_in[10];  const float* b_d2 = (const float*)d_in[11];
  const float* w_se1= (const float*)d_in[12];  const float* b_se1= (const float*)d_in[13];
  const float* w_se2= (const float*)d_in[14];  const float* b_se2= (const float*)d_in[15];
  const float* w_m1 = (const float*)d_in[16];  const float* b_m1 = (const float*)d_in[17];
  const float* w_m2 = (const float*)d_in[18];  const float* b_m2 = (const float*)d_in[19];
  float* out = (float*)d_out;

  // ---- workspace layout (floats, 256B aligned) ----
  float* ws = (float*)d_ws;
  size_t off = 0;
  auto alloc = [&](size_t nfl) { float* p = ws + off; off += (nfl + 63) & ~(size_t)63; return p; };
  float* featN = alloc((size_t)NPIX * CCH);   // reused as x after z-conv consumes it
  float* zf    = alloc((size_t)NPIX * DDIM);
  float* qN    = alloc((size_t)NPIX * DDIM);
  float* h1    = alloc((size_t)NPIX * DDIM);
  float* hA    = alloc((size_t)NPIX * C3);    // logits (N*K) first, then d1 out, then hOut
  float* hB    = alloc((size_t)NPIX * C3);
  float* wm1p  = alloc(4 * 64 * 64);
  float* wm2p  = alloc(13 * 512 * 64);
  float* ek2   = alloc(512);
  float* meanb = alloc(1024);
  float* rsb   = alloc(1024);
  float* sbuf  = alloc(NBATCH * CCH);
  float* s2buf = alloc(NBATCH * CCH);
  float* partials = alloc(16384);
  float* zrow  = alloc(128);
  float* accum = alloc(64);
  int* idx    = (int*)alloc(NPIX);
  int* counts = (int*)alloc(512);
  (void)in_sizes; (void)n_in; (void)out_size; (void)ws_size;

  hipMemsetAsync(counts, 0, 512 * sizeof(int), stream);
  hipMemsetAsync(zrow, 0, 128 * sizeof(float), stream);

  // ---- stage 0: layout + weight prep ----
  k_nchw2nhwc<<<65536, 256, 0, stream>>>(feature, featN);
  k_pack_wm1<<<64, 256, 0, stream>>>(w_m1, wm1p);
  k_pack_wm2<<<1664, 256, 0, stream>>>(w_m2, wm2p);
  k_ek2<<<2, 256, 0, stream>>>(emb, ek2);

  // ---- stage 1: z = conv1x1(feature) ; VQ ----
  k_gemm<CCH, DDIM, false><<<1024, 128, 0, stream>>>(featN, w_q, b_q, zf, nullptr, nullptr);
  k_vq<<<1024, 128, 0, stream>>>(zf, emb, ek2, idx);
  k_hist<<<256, 256, 0, stream>>>(idx, counts);
  k_gather_commit<<<16384, 256, 0, stream>>>(zf, emb, idx, qN, partials);
  k_combine<<<1, 256, 0, stream>>>(partials, 16384, 1.f / ((float)NPIX * DDIM), accum + 0);

  // ---- stage 2: PixelCNN prior (logits aliased into hA) ----
  k_mconv<4, DDIM><<<1024, 128, 0, stream>>>(qN, wm1p, b_m1, h1, zrow);
  k_bnstats<<<DDIM, 256, 0, stream>>>(h1, DDIM, meanb, rsb);
  k_bnapply<<<16384, 256, 0, stream>>>(h1, DDIM, meanb, rsb, 1);
  k_mconv<13, KCB><<<8192, 128, 0, stream>>>(h1, wm2p, b_m2, hA, zrow);
  k_nll<<<8192, 256, 0, stream>>>(hA, idx, partials);
  k_combine<<<1, 256, 0, stream>>>(partials, 8192, 1.f / (float)NPIX, accum + 1);

  // ---- stage 3: decoder cell (x aliased into featN) ----
  k_gemm<DDIM, CCH, false><<<4096, 128, 0, stream>>>(qN, w_c, b_c, featN, nullptr, nullptr);
  k_bnstats<<<CCH, 256, 0, stream>>>(featN, CCH, meanb, rsb);
  k_gemm<CCH, C3, true><<<12288, 128, 0, stream>>>(featN, w_d1, b_d1, hA, meanb, rsb);
  k_bnstats<<<C3, 256, 0, stream>>>(hA, C3, meanb, rsb);
  k_bnapply<<<196608, 256, 0, stream>>>(hA, C3, meanb, rsb, 1);
  k_dw<<<49152, 256, 0, stream>>>(hA, w_dw, b_dw, hB);
  k_bnstats<<<C3, 256, 0, stream>>>(hB, C3, meanb, rsb);
  k_bnapply<<<196608, 256, 0, stream>>>(hB, C3, meanb, rsb, 1);
  k_gemm<C3, CCH, false><<<4096, 128, 0, stream>>>(hB, w_d2, b_d2, hA, nullptr, nullptr);
  k_bnstats<<<CCH, 256, 0, stream>>>(hA, CCH, meanb, rsb);
  k_bnapply<<<65536, 256, 0, stream>>>(hA, CCH, meanb, rsb, 0);

  // ---- stage 4: squeeze-excite + residual + scalars ----
  k_squeeze<<<4096, 128, 0, stream>>>(hA, sbuf);
  k_se<<<1, 256, 0, stream>>>(sbuf, w_se1, b_se1, w_se2, b_se2, s2buf);
  k_final<<<65536, 256, 0, stream>>>(featN, hA, s2buf, out);
  k_scalars<<<1, 64, 0, stream>>>(accum, counts, out + (size_t)NPIX * CCH);
}